// AtomEdgeInteraction_46840913330368
// MI455X (gfx1250) — compile-verified
//
#include <hip/hip_runtime.h>
#include <hip/hip_bf16.h>

#define N_NODES 100000
#define N_EDGES 3200000
#define IN_F    128
#define EDGE_F  16
#define K_TOT   144          // IN_F + EDGE_F
#define OUT_F   128
#define EDGE_IMPORTANCE 1.0f

typedef float v2f __attribute__((ext_vector_type(2)));
typedef float v8f __attribute__((ext_vector_type(8)));

// ---------------------------------------------------------------------------
// Kernel 0: zero the accumulator workspace (agg[N_NODES][K_TOT] + cnt[N_NODES])
// ---------------------------------------------------------------------------
__global__ void zero_ws_kernel(float4* __restrict__ ws, long n4) {
    long i = (long)blockIdx.x * blockDim.x + threadIdx.x;
    long stride = (long)gridDim.x * blockDim.x;
    float4 z = make_float4(0.f, 0.f, 0.f, 0.f);
    for (; i < n4; i += stride) ws[i] = z;
}

// ---------------------------------------------------------------------------
// Kernel 1: per-edge scatter-add.  One wave (32 lanes) per edge.
//   agg[col] += [ x[row] (128f) ; edge_attr[e] (16f) ] ;  cnt[col] += 1
// Edge index is wave-uniform -> scalar loads for row/col.
// ---------------------------------------------------------------------------
__global__ void scatter_kernel(const float* __restrict__ x,
                               const int*   __restrict__ ei,
                               const float* __restrict__ ea,
                               float* __restrict__ agg,
                               float* __restrict__ cnt) {
    const int lane = threadIdx.x & 31;
    const long e = (long)blockIdx.x * (blockDim.x >> 5) + (threadIdx.x >> 5);
    if (e >= N_EDGES) return;

    const int row = ei[e];             // edge_index[0][e]
    const int col = ei[N_EDGES + e];   // edge_index[1][e]

    // 128-bit coalesced gather of the source-node row (fits in L2: 51 MB)
    const float4 v = ((const float4*)(x + (long)row * IN_F))[lane];

    float* dst = agg + (long)col * K_TOT;
    const int f = lane * 4;
    atomicAdd(dst + f + 0, v.x);
    atomicAdd(dst + f + 1, v.y);
    atomicAdd(dst + f + 2, v.z);
    atomicAdd(dst + f + 3, v.w);

    if (lane < EDGE_F) {
        atomicAdd(dst + IN_F + lane, ea[e * EDGE_F + lane] * EDGE_IMPORTANCE);
    } else if (lane == EDGE_F) {
        atomicAdd(cnt + col, 1.0f);
    }
}

// ---------------------------------------------------------------------------
// Kernel 2: node GEMM via V_WMMA_F32_16X16X4_F32 + fused bias/mean epilogue.
//   out[n,o] = ( sum_k agg[n,k]*W[o,k] + cnt[n]*b[o] ) / max(cnt[n],1)
// Block = 256 threads = 8 waves; block owns 16 node rows (M), each wave owns
// one 16-wide output tile (N).  100000 = 6250 * 16 exactly (no tail).
// ---------------------------------------------------------------------------
__global__ void wmma_gemm_kernel(const float* __restrict__ agg,
                                 const float* __restrict__ cnt,
                                 const float* __restrict__ W,
                                 const float* __restrict__ b,
                                 float* __restrict__ out) {
    const int wave = threadIdx.x >> 5;   // 0..7 -> N tile
    const int lane = threadIdx.x & 31;
    const int half = lane >> 4;          // 0: K pair {0,1}, 1: K pair {2,3}
    const int l16  = lane & 15;

    const long mbase = (long)blockIdx.x * 16;
    const int  o     = wave * 16 + l16;  // this lane's output column

    // A: 16x4 fp32 tile of agg; lane(l16) = row M, vgpr pair = K (+2 if half)
    const float* arow = agg + (mbase + l16) * (long)K_TOT;
    // B: 4x16 fp32 tile of W^T; lane(l16) = col N (output o), same K mapping
    const float* wrow = W + (long)o * K_TOT;

    v8f c = {};
#pragma unroll
    for (int k0 = 0; k0 < K_TOT; k0 += 4) {
        const int k = k0 + 2 * half;
        v2f a, bm;
        a.x  = arow[k + 0];
        a.y  = arow[k + 1];
        bm.x = wrow[k + 0];
        bm.y = wrow[k + 1];
        c = __builtin_amdgcn_wmma_f32_16x16x4_f32(
                /*neg_a=*/false, a, /*neg_b=*/false, bm,
                /*c_mod=*/(short)0, c, /*reuse_a=*/false, /*reuse_b=*/false);
    }

    // Epilogue: C/D layout -> vgpr r, row M = r + 8*half, col N = l16 (== o)
    const float bias = b[o];
#pragma unroll
    for (int r = 0; r < 8; ++r) {
        const long m = mbase + r + 8 * half;
        const float cn = cnt[m];
        const float denom = fmaxf(cn, 1.0f);
        out[m * OUT_F + o] = (c[r] + cn * bias) / denom;
    }
}

// ---------------------------------------------------------------------------
extern "C" void kernel_launch(void* const* d_in, const int* in_sizes, int n_in,
                              void* d_out, int out_size, void* d_ws, size_t ws_size,
                              hipStream_t stream) {
    const float* x  = (const float*)d_in[0];   // (N_NODES, IN_F)
    const int*   ei = (const int*)  d_in[1];   // (2, N_EDGES)
    const float* ea = (const float*)d_in[2];   // (N_EDGES, EDGE_F)
    const float* W  = (const float*)d_in[3];   // (OUT_F, K_TOT)
    const float* b  = (const float*)d_in[4];   // (OUT_F,)
    float* out = (float*)d_out;

    float* agg = (float*)d_ws;                             // N_NODES * K_TOT
    float* cnt = agg + (size_t)N_NODES * K_TOT;            // N_NODES

    // Phase 0: zero accumulators (14.5M floats, /4 = float4 count)
    const long n4 = ((long)N_NODES * K_TOT + N_NODES) / 4;
    zero_ws_kernel<<<2048, 256, 0, stream>>>((float4*)d_ws, n4);

    // Phase 1: scatter-add.  8 edges per 256-thread block (1 wave per edge).
    const int edges_per_block = 256 / 32;
    const int nblk1 = (N_EDGES + edges_per_block - 1) / edges_per_block; // 400000
    scatter_kernel<<<nblk1, 256, 0, stream>>>(x, ei, ea, agg, cnt);

    // Phase 2: fp32 WMMA GEMM + fused mean/bias.  6250 blocks x 8 waves.
    const int nblk2 = N_NODES / 16;  // 6250
    wmma_gemm_kernel<<<nblk2, 256, 0, stream>>>(agg, cnt, W, b, out);
}